// GNN_node_21105469293089
// MI455X (gfx1250) — compile-verified
//
#include <hip/hip_runtime.h>
#include <hip/hip_bf16.h>

typedef __attribute__((ext_vector_type(2))) float v2f;
typedef __attribute__((ext_vector_type(8))) float v8f;

#define N_NODES_C 100000
#define N_EDGES_C 1600000
#define DIM 128

// ---------------------------------------------------------------------------
// Degree / norm kernels
// ---------------------------------------------------------------------------
__global__ void zero_norm_kernel(float* a, float* b, int n) {
    int i = blockIdx.x * blockDim.x + threadIdx.x;
    if (i < n) { a[i] = 0.0f; b[i] = 0.0f; }
}

__global__ void degree_kernel(const int* __restrict__ src, const int* __restrict__ dst,
                              float* odeg, float* ideg, int n_edges) {
    int e = blockIdx.x * blockDim.x + threadIdx.x;
    if (e < n_edges) {
        atomicAdd(&odeg[src[e]], 1.0f);
        atomicAdd(&ideg[dst[e]], 1.0f);
    }
}

__global__ void finalize_norm_kernel(float* a, float* b, int n) {
    int i = blockIdx.x * blockDim.x + threadIdx.x;
    if (i < n) {
        a[i] = rsqrtf(fmaxf(a[i], 1.0f));
        b[i] = rsqrtf(fmaxf(b[i], 1.0f));
    }
}

// ---------------------------------------------------------------------------
// Zero the aggregate buffer (float4 granularity).
// ---------------------------------------------------------------------------
__global__ void zero_buf_kernel(float4* __restrict__ m, int n_f4) {
    int idx = blockIdx.x * blockDim.x + threadIdx.x;
    if (idx < n_f4) m[idx] = make_float4(0.f, 0.f, 0.f, 0.f);
}

// ---------------------------------------------------------------------------
// Fused SpMM scatter: one wave32 per edge.  Each lane gathers float4 of
// h[src], scales by norm_src[src] in-register (saves a whole 150MB/layer
// pre-scale pass), and scatters with 4 global_atomic_add_f32 into m[dst].
// Entire working set (~115MB) is L2-resident on the 192MB L2.
// ---------------------------------------------------------------------------
__global__ void spmm_kernel(const float* __restrict__ h, const float* __restrict__ ns,
                            const int* __restrict__ src, const int* __restrict__ dst,
                            float* __restrict__ m, int n_edges) {
    int gid  = blockIdx.x * blockDim.x + threadIdx.x;
    int edge = gid >> 5;
    int lane = gid & 31;
    if (edge >= n_edges) return;
    int s = src[edge];
    int d = dst[edge];
    float scale = ns[s];
    float4 v = ((const float4*)(h + (size_t)s * DIM))[lane];
    v.x *= scale; v.y *= scale; v.z *= scale; v.w *= scale;
    float* out = m + (size_t)d * DIM + lane * 4;
    atomicAdd(out + 0, v.x);
    atomicAdd(out + 1, v.y);
    atomicAdd(out + 2, v.z);
    atomicAdd(out + 3, v.w);
}

// ---------------------------------------------------------------------------
// GEMM: out[i,:] = (m[i,:] * norm_dst[i]) @ W + bias, optional ReLU.
// Full-precision f32 matrix pipe: V_WMMA_F32_16X16X4_F32.
// Block = 256 threads = 8 waves; each wave owns a 16-node row tile and all
// 128 output columns (8 accumulators of 16x16).  W staged in LDS (64 KB).
//
// A 16x4 f32 layout : lanes 0-15 hold M=lane, VGPR{0,1}=K{0,1};
//                     lanes 16-31 hold M=lane-16, VGPR{0,1}=K{2,3}.
// B 4x16 f32 layout : VGPR0: K=0 (lanes 0-15), K=2 (lanes 16-31);
//                     VGPR1: K=1 / K=3; N = lane%16.
// C/D 16x16 layout  : VGPR r: M = r + 8*(lane/16), N = lane%16.
// ---------------------------------------------------------------------------
__global__ __launch_bounds__(256) void gemm_wmma_kernel(
        const float* __restrict__ M, const float* __restrict__ nd,
        const float* __restrict__ W, const float* __restrict__ bias,
        float* __restrict__ out, int n_nodes, int do_relu) {
    __shared__ float sW[DIM * DIM];   // 64 KB

    int tid = threadIdx.x;
    // Stage W into LDS (row-major [k][n]).
    for (int i = tid; i < DIM * DIM / 4; i += 256)
        ((float4*)sW)[i] = ((const float4*)W)[i];
    __syncthreads();

    int wave = tid >> 5;
    int lane = tid & 31;
    int rlane = lane & 15;            // row within tile (A) / column (B,C,D)
    int half  = lane >> 4;            // 0/1 -> K-pair select, M+8 select
    int tile_base = (blockIdx.x * 8 + wave) * 16;

    int row  = tile_base + rlane;
    int rowc = row < n_nodes ? row : (n_nodes - 1);   // clamp, keep EXEC full
    float ndv = nd[rowc];

    v8f acc[8];
#pragma unroll
    for (int t = 0; t < 8; ++t) acc[t] = (v8f){0.f,0.f,0.f,0.f,0.f,0.f,0.f,0.f};

    const float* mrow = M + (size_t)rowc * DIM;
    for (int kb = 0; kb < DIM; kb += 4) {
        int k = kb + 2 * half;
        v2f a;
        a.x = mrow[k + 0] * ndv;
        a.y = mrow[k + 1] * ndv;
#pragma unroll
        for (int t = 0; t < 8; ++t) {
            int n = t * 16 + rlane;
            v2f b;
            b.x = sW[(k + 0) * DIM + n];
            b.y = sW[(k + 1) * DIM + n];
            acc[t] = __builtin_amdgcn_wmma_f32_16x16x4_f32(
                false, a, false, b, (short)0, acc[t], false, false);
        }
    }

    // Epilogue: bias (+ReLU), guarded scattered stores for the row tail.
#pragma unroll
    for (int t = 0; t < 8; ++t) {
        int n = t * 16 + rlane;
        float bv = bias[n];
#pragma unroll
        for (int r = 0; r < 8; ++r) {
            int mrow_idx = tile_base + r + 8 * half;
            float v = acc[t][r] + bv;
            if (do_relu) v = fmaxf(v, 0.0f);
            if (mrow_idx < n_nodes)
                out[(size_t)mrow_idx * DIM + n] = v;
        }
    }
}

// ---------------------------------------------------------------------------
// Host-side orchestration
// ---------------------------------------------------------------------------
extern "C" void kernel_launch(void* const* d_in, const int* in_sizes, int n_in,
                              void* d_out, int out_size, void* d_ws, size_t ws_size,
                              hipStream_t stream) {
    const float* x   = (const float*)d_in[0];
    const int*   src = (const int*)  d_in[1];
    const int*   dst = (const int*)  d_in[2];
    const float* Ws[3] = { (const float*)d_in[3], (const float*)d_in[5], (const float*)d_in[7] };
    const float* bs[3] = { (const float*)d_in[4], (const float*)d_in[6], (const float*)d_in[8] };
    float* out = (float*)d_out;

    const int n_nodes = N_NODES_C;
    const int n_edges = N_EDGES_C;

    // Workspace layout (all 256B-aligned):
    //   [0, 512K)      norm_src (deg reuse)
    //   [512K, 1M)     norm_dst (deg reuse)
    //   [1M, 1M+52M)   bufA : h (layer outputs)
    //   [57M, 57M+52M) bufB : m aggregate
    char* ws = (char*)d_ws;
    float* norm_src = (float*)(ws);
    float* norm_dst = (float*)(ws + (512u << 10));
    float* bufA     = (float*)(ws + (1u  << 20));
    float* bufB     = (float*)(ws + (57ull << 20));

    const int B = 256;

    // Degrees -> norms
    zero_norm_kernel<<<(n_nodes + B - 1) / B, B, 0, stream>>>(norm_src, norm_dst, n_nodes);
    degree_kernel<<<(n_edges + B - 1) / B, B, 0, stream>>>(src, dst, norm_src, norm_dst, n_edges);
    finalize_norm_kernel<<<(n_nodes + B - 1) / B, B, 0, stream>>>(norm_src, norm_dst, n_nodes);

    const int n_f4         = n_nodes * (DIM / 4);                     // 3.2M float4
    const int zero_blocks  = (n_f4 + B - 1) / B;                      // 12500
    const int spmm_blocks  = ((n_edges * 32) + B - 1) / B;            // 200000
    const int gemm_blocks  = (n_nodes + 127) / 128;                   // 782

    const float* hin = x;
    for (int l = 0; l < 3; ++l) {
        // m = 0
        zero_buf_kernel<<<zero_blocks, B, 0, stream>>>((float4*)bufB, n_f4);
        // m[dst] += h[src] * norm_src[src]   (scale fused into gather)
        spmm_kernel<<<spmm_blocks, B, 0, stream>>>(hin, norm_src, src, dst, bufB, n_edges);
        // h_out = (m * norm_dst) @ W + b (+ReLU except last)
        float* hout = (l == 2) ? out : bufA;
        gemm_wmma_kernel<<<gemm_blocks, B, 0, stream>>>(bufB, norm_dst, Ws[l], bs[l],
                                                        hout, n_nodes, (l != 2) ? 1 : 0);
        hin = bufA;
    }
}